// fmmLayer_59906203845033
// MI455X (gfx1250) — compile-verified
//
#include <hip/hip_runtime.h>

typedef __attribute__((ext_vector_type(2))) float v2f;
typedef __attribute__((ext_vector_type(8))) float v8f;

#define NTOT 512          // NCELLS * NP
#define BLOCK 256         // 8 waves (wave32)

// Per-pair features: {1/r, 1/sqrt(r), 1/r^2, i0e(r)}, masked to 0 on diagonal.
// Uses hardware v_rcp_f32 / v_rsq_f32 / v_exp_f32 (no correctness harness;
// A&S 9.8.1/9.8.2 polynomial for i0e matches jax i0e to ~1e-7).
__device__ __forceinline__ void pair_feats(float ri, float rj, bool diag, float f[4]) {
    float d = __builtin_fabsf(rj - ri);
    float x = diag ? 1.0f : d;                     // safe diagonal value
    float inv  = __builtin_amdgcn_rcpf(x);         // 1/r
    float rs   = __builtin_amdgcn_rsqf(x);         // 1/sqrt(r)
    float inv2 = inv * inv;                        // 1/r^2

    // i0e(x) = exp(-x) * I0(x), x >= 0
    // small branch: t2 = (x/3.75)^2
    float t  = x * 0.2666666666666667f;
    float t2 = t * t;
    float ps = (((((0.0045813f * t2 + 0.0360768f) * t2 + 0.2659732f) * t2
                 + 1.2067492f) * t2 + 3.0899424f) * t2 + 3.5156229f) * t2 + 1.0f;
    float es = __builtin_amdgcn_exp2f(x * -1.4426950408889634f);   // exp(-x)
    float i0e_small = ps * es;
    // large branch: u = 3.75/x
    float u = 3.75f * inv;
    float pl = (((((((0.00392377f * u - 0.01647633f) * u + 0.02635537f) * u
                   - 0.02057706f) * u + 0.00916281f) * u - 0.00157565f) * u
                 + 0.00225319f) * u + 0.01328592f) * u + 0.39894228f;
    float i0e_large = pl * rs;                      // * 1/sqrt(x)
    float i0e = (x <= 3.75f) ? i0e_small : i0e_large;

    float m = diag ? 0.0f : 1.0f;
    f[0] = inv  * m;
    f[1] = rs   * m;
    f[2] = inv2 * m;
    f[3] = i0e  * m;
}

__global__ __launch_bounds__(BLOCK)
void fmm_wmma_kernel(const float* __restrict__ Rin,
                     const float* __restrict__ std4,
                     const float* __restrict__ bias4,
                     float* __restrict__ out)
{
    __shared__ float Rl[NTOT];

    const int tid  = threadIdx.x;
    const int b    = blockIdx.x >> 2;     // 4 blocks per batch
    const int grp  = blockIdx.x & 3;
    const int wave = tid >> 5;
    const int lane = tid & 31;

    // Stage this batch's 512 coordinates into LDS (2 KB).
    const float* Rb = Rin + b * NTOT;
    Rl[tid]       = Rb[tid];
    Rl[tid + 256] = Rb[tid + 256];
    __syncthreads();

    // Each wave owns a 16-row i-tile.
    const int itile = grp * 8 + wave;     // 0..31
    const int ibase = itile * 16;
    const int m     = lane & 15;          // A-matrix row M = lane % 16
    const int gi    = ibase + m;
    const float ri  = Rl[gi];
    // A-operand K mapping: lanes 0-15 -> K=0,1 ; lanes 16-31 -> K=2,3
    const int joff  = (lane >> 4) << 1;

    v8f acc0 = {}, acc1 = {}, acc2 = {}, acc3 = {};
    const v2f ones = {1.0f, 1.0f};        // B = all-ones 4x16 => D[m,n] = sum_k A[m,k] + C

    #pragma unroll 4
    for (int j0 = 0; j0 < NTOT; j0 += 4) {
        const int jA = j0 + joff;
        float fA[4], fB[4];
        pair_feats(ri, Rl[jA],     gi == jA,       fA);
        pair_feats(ri, Rl[jA + 1], gi == (jA + 1), fB);
        v2f a0 = {fA[0], fB[0]};
        v2f a1 = {fA[1], fB[1]};
        v2f a2 = {fA[2], fB[2]};
        v2f a3 = {fA[3], fB[3]};
        // (neg_a, A, neg_b, B, c_mod, C, reuse_a, reuse_b)
        acc0 = __builtin_amdgcn_wmma_f32_16x16x4_f32(false, a0, false, ones, (short)0, acc0, false, false);
        acc1 = __builtin_amdgcn_wmma_f32_16x16x4_f32(false, a1, false, ones, (short)0, acc1, false, false);
        acc2 = __builtin_amdgcn_wmma_f32_16x16x4_f32(false, a2, false, ones, (short)0, acc2, false, false);
        acc3 = __builtin_amdgcn_wmma_f32_16x16x4_f32(false, a3, false, ones, (short)0, acc3, false, false);
    }

    // Epilogue: every column n of each accumulator holds the same sum.
    // C/D layout: lane 0 (col 0) has rows 0..7 in v0..7; lane 16 has rows 8..15.
    // out[b,i,k] = |std_k| * (S_ik - bias_k * (N-1))
    if (m == 0) {
        const float c511 = (float)(NTOT - 1);
        const float s0 = __builtin_fabsf(std4[0]), b0 = bias4[0] * c511;
        const float s1 = __builtin_fabsf(std4[1]), b1 = bias4[1] * c511;
        const float s2 = __builtin_fabsf(std4[2]), b2 = bias4[2] * c511;
        const float s3 = __builtin_fabsf(std4[3]), b3 = bias4[3] * c511;
        const int mbase = (lane >> 4) << 3;       // 0 or 8
        float4* outv = reinterpret_cast<float4*>(out);
        #pragma unroll
        for (int v = 0; v < 8; ++v) {
            const int i = ibase + mbase + v;
            float4 o;
            o.x = s0 * (acc0[v] - b0);
            o.y = s1 * (acc1[v] - b1);
            o.z = s2 * (acc2[v] - b2);
            o.w = s3 * (acc3[v] - b3);
            outv[b * NTOT + i] = o;
        }
    }
}

extern "C" void kernel_launch(void* const* d_in, const int* in_sizes, int n_in,
                              void* d_out, int out_size, void* d_ws, size_t ws_size,
                              hipStream_t stream) {
    const float* Rin   = (const float*)d_in[0];   // [B, 16, 32] f32
    const float* std4  = (const float*)d_in[1];   // [4] f32
    const float* bias4 = (const float*)d_in[2];   // [4] f32
    float* out = (float*)d_out;                   // [B, 512, 4] f32

    const int B = in_sizes[0] / NTOT;             // 128
    dim3 grid(B * 4), block(BLOCK);
    hipLaunchKernelGGL(fmm_wmma_kernel, grid, block, 0, stream,
                       Rin, std4, bias4, out);
}